// ODETexture_34737695490060
// MI455X (gfx1250) — compile-verified
//
#include <hip/hip_runtime.h>

#define NPTS 32768
#define DDIM 3
#define NDSZ (NPTS * DDIM)     // 98304
#define HDIM 64
#define T_END_F 1.0f
#define WAVES_PER_BLK 8

typedef __attribute__((ext_vector_type(2))) float v2f;
typedef __attribute__((ext_vector_type(8))) float v8f;

static __device__ __forceinline__ v8f wmma_f32_k4(v2f a, v2f b, v8f c) {
  // V_WMMA_F32_16X16X4_F32 : D(16x16 f32) = A(16x4 f32) * B(4x16 f32) + C
  return __builtin_amdgcn_wmma_f32_16x16x4_f32(false, a, false, b, (short)0, c,
                                               false, false);
}

static __device__ __forceinline__ v8f splat8(float v) {
  v8f r;
#pragma unroll
  for (int i = 0; i < 8; ++i) r[i] = v;
  return r;
}

static __device__ __forceinline__ float tanh_fast(float x) {
  // tanh(x) = sign(x) * (1 - e) / (1 + e),  e = exp(-2|x|); rcp via v_rcp_f32
  float ax = fabsf(x);
  float e  = __expf(-2.0f * ax);
  float r  = (1.0f - e) * __builtin_amdgcn_rcpf(1.0f + e);
  return copysignf(r, x);
}

// ---------------------------------------------------------------------------
// Stage kernel: computes k_stage = f(t + ci*dt_c, y + dt_c * sum_j a_j * k_j)
// One wave (32 lanes) handles a 16-row tile; 8 waves / block.
// ---------------------------------------------------------------------------
__global__ __launch_bounds__(32 * WAVES_PER_BLK) void stage_kernel(
    const float* __restrict__ y, float* ks, const float* __restrict__ ctrl,
    const float* __restrict__ W1, const float* __restrict__ b1,
    const float* __restrict__ W2, const float* __restrict__ b2,
    const float* __restrict__ W3, const float* __restrict__ b3, int stage,
    int nacc, float a0, float a1, float a2, float a3, float a4, float a5,
    float ci) {
  __shared__ float sW1[4 * 64];
  __shared__ float sW2[64 * 64];
  __shared__ float sW3[64 * 16];  // zero-padded columns 3..15
  __shared__ float sB1[64];
  __shared__ float sB2[64];
  __shared__ float sB3[16];
  __shared__ float sH[WAVES_PER_BLK][16 * 68];  // per-wave tile, stride 68

  const int tid  = threadIdx.x;
  const int nthr = 32 * WAVES_PER_BLK;
  for (int i = tid; i < 4 * 64; i += nthr) sW1[i] = W1[i];
  for (int i = tid; i < 64 * 64; i += nthr) sW2[i] = W2[i];
  for (int i = tid; i < 64 * 16; i += nthr) {
    int n = i & 15;
    sW3[i] = (n < DDIM) ? W3[(i >> 4) * DDIM + n] : 0.0f;
  }
  if (tid < 64) {
    sB1[tid] = b1[tid];
    sB2[tid] = b2[tid];
  }
  if (tid < 16) sB3[tid] = (tid < DDIM) ? b3[tid] : 0.0f;
  __syncthreads();

  const float tcur = ctrl[0];
  const float dtc  = ctrl[2];
  const float ti   = tcur + ci * dtc;

  const int wave = tid >> 5;
  const int lane = tid & 31;
  const int m    = lane & 15;  // row within tile (A), col within tile (B/C)
  const int half = lane >> 4;  // 0 or 1 -> K split
  const int tile = blockIdx.x * WAVES_PER_BLK + wave;
  const int row  = tile * 16 + m;

  // ---- build layer-1 A fragment: x = [yi(3), t], K split across halves ----
  const int c0 = half * 2;                             // component 0 or 2
  const int c1 = (c0 + 1 < DDIM) ? c0 + 1 : DDIM - 1;  // clamp (lane>=16 -> t)
  const size_t base = (size_t)row * DDIM;
  float x0 = y[base + c0];
  float x1 = y[base + c1];
  const float av[6] = {a0, a1, a2, a3, a4, a5};
#pragma unroll
  for (int j = 0; j < 6; ++j) {
    if (j < nacc) {
      const float* kj = ks + (size_t)j * NDSZ;
      float w = dtc * av[j];
      x0 += w * kj[base + c0];
      x1 += w * kj[base + c1];
    }
  }
  if (half) x1 = ti;  // K=3 column is the time input

  v2f afrag;
  afrag.x = x0;
  afrag.y = x1;

  float* h = &sH[wave][0];

  // ---- layer 1: h1 = tanh(x @ W1 + b1), 16x4 @ 4x64 ----
  v8f acc[4];
  {
    const int kb = half * 2;
#pragma unroll
    for (int nb = 0; nb < 4; ++nb) {
      v8f c = splat8(sB1[nb * 16 + m]);
      v2f bf;
      bf.x = sW1[(kb + 0) * 64 + nb * 16 + m];
      bf.y = sW1[(kb + 1) * 64 + nb * 16 + m];
      acc[nb] = wmma_f32_k4(afrag, bf, c);
    }
  }
#pragma unroll
  for (int nb = 0; nb < 4; ++nb)
#pragma unroll
    for (int r = 0; r < 8; ++r) {
      int M = r + half * 8;
      h[M * 68 + nb * 16 + m] = tanh_fast(acc[nb][r]);
    }

  // ---- layer 2: h2 = tanh(h1 @ W2 + b2), 16x64 @ 64x64 ----
#pragma unroll
  for (int nb = 0; nb < 4; ++nb) acc[nb] = splat8(sB2[nb * 16 + m]);
#pragma unroll
  for (int ks4 = 0; ks4 < 16; ++ks4) {
    const int kb = ks4 * 4 + half * 2;
    v2f a2;
    a2.x = h[m * 68 + kb + 0];
    a2.y = h[m * 68 + kb + 1];
#pragma unroll
    for (int nb = 0; nb < 4; ++nb) {
      v2f bf;
      bf.x = sW2[(kb + 0) * 64 + nb * 16 + m];
      bf.y = sW2[(kb + 1) * 64 + nb * 16 + m];
      acc[nb] = wmma_f32_k4(a2, bf, acc[nb]);
    }
  }
#pragma unroll
  for (int nb = 0; nb < 4; ++nb)
#pragma unroll
    for (int r = 0; r < 8; ++r) {
      int M = r + half * 8;
      h[M * 68 + nb * 16 + m] = tanh_fast(acc[nb][r]);
    }

  // ---- layer 3: out = h2 @ W3 + b3, 16x64 @ 64x16(padded) ----
  {
    v8f acc3 = splat8(sB3[m]);
#pragma unroll
    for (int ks4 = 0; ks4 < 16; ++ks4) {
      const int kb = ks4 * 4 + half * 2;
      v2f a3;
      a3.x = h[m * 68 + kb + 0];
      a3.y = h[m * 68 + kb + 1];
      v2f bf;
      bf.x = sW3[(kb + 0) * 16 + m];
      bf.y = sW3[(kb + 1) * 16 + m];
      acc3 = wmma_f32_k4(a3, bf, acc3);
    }
    float* kout = ks + (size_t)stage * NDSZ;
    if (m < DDIM) {
#pragma unroll
      for (int r = 0; r < 8; ++r) {
        int M = r + half * 8;
        kout[(size_t)(tile * 16 + M) * DDIM + m] = acc3[r];
      }
    }
  }
}

// ---------------------------------------------------------------------------
// Combine: y_new, y_err, deterministic per-block sum of (y_err/tol)^2
// ---------------------------------------------------------------------------
__global__ __launch_bounds__(256) void combine_kernel(
    const float* __restrict__ y, const float* __restrict__ ks,
    const float* __restrict__ ctrl, float* __restrict__ ynew,
    float* __restrict__ partials, float bb0, float bb1, float bb2, float bb3,
    float bb4, float bb5, float bb6, float ee0, float ee1, float ee2,
    float ee3, float ee4, float ee5, float ee6) {
  __shared__ float red[256];
  const int idx   = blockIdx.x * 256 + threadIdx.x;
  const float dtc = ctrl[2];
  const float bs[7] = {bb0, bb1, bb2, bb3, bb4, bb5, bb6};
  const float es[7] = {ee0, ee1, ee2, ee3, ee4, ee5, ee6};
  float yv = y[idx];
  float yn = yv;
  float ye = 0.0f;
#pragma unroll
  for (int i = 0; i < 7; ++i) {
    float kv = ks[(size_t)i * NDSZ + idx];
    yn += (dtc * bs[i]) * kv;
    ye += (dtc * es[i]) * kv;
  }
  ynew[idx] = yn;
  float tol = 1e-5f + 1e-5f * fmaxf(fabsf(yv), fabsf(yn));
  float r   = ye / tol;  // keep IEEE division: feeds accept/reject decision
  red[threadIdx.x] = r * r;
  __syncthreads();
  for (int s = 128; s > 0; s >>= 1) {
    if (threadIdx.x < s) red[threadIdx.x] += red[threadIdx.x + s];
    __syncthreads();
  }
  if (threadIdx.x == 0) partials[blockIdx.x] = red[0];
}

// ---------------------------------------------------------------------------
// Control: (post) accept/step-size logic, then (pre) dt_c/active for next step
// ctrl layout: [0]=t [1]=dt [2]=dt_c [3]=active [4]=upd
// ---------------------------------------------------------------------------
__global__ void ctrl_kernel(float* ctrl, const float* __restrict__ partials,
                            int npart, int first) {
  if (first) {
    ctrl[0] = 0.0f;
    ctrl[1] = 0.05f;
    ctrl[4] = 0.0f;
  } else {
    float sum = 0.0f;
    for (int i = 0; i < npart; ++i) sum += partials[i];
    float err = sqrtf(sum / (float)NDSZ);
    err = fmaxf(err, 1e-10f);
    int accept   = (err <= 1.0f) ? 1 : 0;
    float factor = fminf(fmaxf(0.9f * powf(err, -0.2f), 0.2f), 10.0f);
    float t = ctrl[0], dt = ctrl[1], dtc = ctrl[2];
    int active = (ctrl[3] != 0.0f) ? 1 : 0;
    int upd    = active && accept;
    if (upd) t += dtc;
    if (active) dt = dtc * factor;
    ctrl[0] = t;
    ctrl[1] = dt;
    ctrl[4] = upd ? 1.0f : 0.0f;
  }
  // pre-phase for next step
  float t = ctrl[0], dt = ctrl[1];
  float remaining = T_END_F - t;
  float active = (remaining > 1e-9f) ? 1.0f : 0.0f;
  float dtc = (active != 0.0f) ? fminf(dt, remaining) : 1.0f;
  ctrl[2] = dtc;
  ctrl[3] = active;
}

__global__ __launch_bounds__(256) void apply_kernel(
    float* __restrict__ y, const float* __restrict__ ynew,
    const float* __restrict__ ctrl) {
  if (ctrl[4] != 0.0f) {
    int idx = blockIdx.x * 256 + threadIdx.x;
    y[idx] = ynew[idx];
  }
}

__global__ __launch_bounds__(256) void copy_kernel(
    float* __restrict__ dst, const float* __restrict__ src) {
  int idx = blockIdx.x * 256 + threadIdx.x;
  dst[idx] = src[idx];
}

// ---------------------------------------------------------------------------
extern "C" void kernel_launch(void* const* d_in, const int* in_sizes, int n_in,
                              void* d_out, int out_size, void* d_ws,
                              size_t ws_size, hipStream_t stream) {
  const float* y0 = (const float*)d_in[0];
  const float* W1 = (const float*)d_in[1];
  const float* b1 = (const float*)d_in[2];
  const float* W2 = (const float*)d_in[3];
  const float* b2 = (const float*)d_in[4];
  const float* W3 = (const float*)d_in[5];
  const float* b3 = (const float*)d_in[6];

  float* ws       = (float*)d_ws;
  float* ctrl     = ws;            // 8 floats
  float* partials = ws + 8;        // 384 floats
  float* ys       = ws + 512;      // NDSZ
  float* ynew     = ys + NDSZ;     // NDSZ
  float* ks       = ynew + NDSZ;   // 7 * NDSZ

  static const double Ad[7][6] = {
      {0, 0, 0, 0, 0, 0},
      {1.0 / 5, 0, 0, 0, 0, 0},
      {3.0 / 40, 9.0 / 40, 0, 0, 0, 0},
      {44.0 / 45, -56.0 / 15, 32.0 / 9, 0, 0, 0},
      {19372.0 / 6561, -25360.0 / 2187, 64448.0 / 6561, -212.0 / 729, 0, 0},
      {9017.0 / 3168, -355.0 / 33, 46732.0 / 5247, 49.0 / 176,
       -5103.0 / 18656, 0},
      {35.0 / 384, 0, 500.0 / 1113, 125.0 / 192, -2187.0 / 6784, 11.0 / 84},
  };
  static const double Cd[7] = {0, 0.2, 0.3, 0.8, 8.0 / 9, 1.0, 1.0};
  static const double Bd[7] = {35.0 / 384,      0,         500.0 / 1113,
                               125.0 / 192,     -2187.0 / 6784,
                               11.0 / 84,       0};
  static const double BSd[7] = {5179.0 / 57600,    0,
                                7571.0 / 16695,    393.0 / 640,
                                -92097.0 / 339200, 187.0 / 2100,
                                1.0 / 40};

  const int nblkE = NDSZ / 256;                     // 384 elementwise blocks
  const int nblkS = NPTS / 16 / WAVES_PER_BLK;      // 256 stage blocks

  copy_kernel<<<nblkE, 256, 0, stream>>>(ys, y0);
  ctrl_kernel<<<1, 1, 0, stream>>>(ctrl, partials, nblkE, 1);

  for (int s = 0; s < 24; ++s) {
    for (int i = 0; i < 7; ++i) {
      float a[6];
      for (int j = 0; j < 6; ++j) a[j] = (float)Ad[i][j];
      stage_kernel<<<nblkS, 32 * WAVES_PER_BLK, 0, stream>>>(
          ys, ks, ctrl, W1, b1, W2, b2, W3, b3, i, i, a[0], a[1], a[2], a[3],
          a[4], a[5], (float)Cd[i]);
    }
    combine_kernel<<<nblkE, 256, 0, stream>>>(
        ys, ks, ctrl, ynew, partials, (float)Bd[0], (float)Bd[1], (float)Bd[2],
        (float)Bd[3], (float)Bd[4], (float)Bd[5], (float)Bd[6],
        (float)(Bd[0] - BSd[0]), (float)(Bd[1] - BSd[1]),
        (float)(Bd[2] - BSd[2]), (float)(Bd[3] - BSd[3]),
        (float)(Bd[4] - BSd[4]), (float)(Bd[5] - BSd[5]),
        (float)(Bd[6] - BSd[6]));
    ctrl_kernel<<<1, 1, 0, stream>>>(ctrl, partials, nblkE, 0);
    apply_kernel<<<nblkE, 256, 0, stream>>>(ys, ynew, ctrl);
  }

  copy_kernel<<<nblkE, 256, 0, stream>>>((float*)d_out, ys);
}